// PillarFeatureNet_79791902425223
// MI455X (gfx1250) — compile-verified
//
#include <hip/hip_runtime.h>

// ---------------- problem constants (from reference) ----------------
#define GX_    512
#define GY_    512
#define BATCH_ 4
#define NSEG_  (BATCH_ * GX_ * GY_)   // 1,048,576 pillars
#define CCH_   64
#define VOX    0.2f
#define PCXY   (-51.2f)
#define INV_VOX 5.0f                  // 1/0.2
#define BN_EPS_ 1e-3f

typedef __attribute__((ext_vector_type(2))) float v2f;
typedef __attribute__((ext_vector_type(8))) float v8f;

// ---------------- pass 1: pillar scatter-sum ----------------
__global__ __launch_bounds__(256) void k_pillar_scatter(
    const float* __restrict__ pts, int* __restrict__ seg,
    float* __restrict__ cnt, float* __restrict__ xsum, int n) {
  int i = blockIdx.x * blockDim.x + threadIdx.x;
  int stride = gridDim.x * blockDim.x;
  for (; i < n; i += stride) {
    const float* q = pts + (size_t)i * 5;
    float b = q[0], x = q[1], y = q[2], z = q[3];
    float cx = (x - PCXY) * INV_VOX;
    float cy = (y - PCXY) * INV_VOX;
    bool valid = (cx >= 0.f) && (cx < (float)GX_) && (cy >= 0.f) && (cy < (float)GY_);
    int ix = (int)floorf(cx), iy = (int)floorf(cy);
    int bi = (int)b;
    int s = valid ? ((bi * GX_ + ix) * GY_ + iy) : NSEG_;
    seg[i] = s;
    if (valid) {
      atomicAdd(&cnt[s], 1.0f);
      atomicAdd(&xsum[s * 3 + 0], x);
      atomicAdd(&xsum[s * 3 + 1], y);
      atomicAdd(&xsum[s * 3 + 2], z);
    }
  }
}

// ---------------- per-point 9-dim feature builder ----------------
struct Feats { float x, y, z, r, fcx, fcy, fcz, fnx, fny; int s; };

__device__ __forceinline__ Feats compute_feats(
    const float* __restrict__ pts, const int* __restrict__ seg,
    const float* __restrict__ cnt, const float* __restrict__ xsum,
    int p, int n) {
  Feats f;
  if (p >= n) {
    f.x = f.y = f.z = f.r = f.fcx = f.fcy = f.fcz = f.fnx = f.fny = 0.f;
    f.s = NSEG_;
    return f;
  }
  const float* q = pts + (size_t)p * 5;
  float x = q[1], y = q[2], z = q[3], r = q[4];
  int s = seg[p];
  float c = cnt[s];                           // dump seg: cnt==0 -> mean 0
  float inv = 1.0f / fmaxf(c, 1.0f);
  float mx = xsum[s * 3 + 0] * inv;
  float my = xsum[s * 3 + 1] * inv;
  float mz = xsum[s * 3 + 2] * inv;
  float cx = (x - PCXY) * INV_VOX;
  float cy = (y - PCXY) * INV_VOX;
  float ix = floorf(cx), iy = floorf(cy);
  f.x = x; f.y = y; f.z = z; f.r = r;
  f.fcx = x - mx; f.fcy = y - my; f.fcz = z - mz;
  f.fnx = x - (ix * VOX + 0.5f * VOX + PCXY);
  f.fny = y - (iy * VOX + 0.5f * VOX + PCXY);
  f.s = s;
  return f;
}

// Load W fragments for V_WMMA_F32_16X16X4_F32.
// B (4x16) layout: lane%16 = column n, lane/16 selects K-pair {0,1} / {2,3}.
__device__ __forceinline__ void load_Bfrag(const float* __restrict__ W,
                                           int nl, int khalf, v2f Bf[3][4]) {
#pragma unroll
  for (int kt = 0; kt < 3; ++kt) {
    int k0 = kt * 4 + 2 * khalf;
    int k1 = k0 + 1;
#pragma unroll
    for (int nt = 0; nt < 4; ++nt) {
      int nch = nt * 16 + nl;
      v2f bb;
      bb[0] = (k0 < 9) ? W[k0 * CCH_ + nch] : 0.f;
      bb[1] = (k1 < 9) ? W[k1 * CCH_ + nch] : 0.f;
      Bf[kt][nt] = bb;
    }
  }
}

// A fragments: lane%16 = row m (point), lane/16 selects K-pair.
__device__ __forceinline__ void make_Afrag(const Feats& f, int khalf, v2f A[3]) {
  A[0][0] = khalf ? f.z   : f.x;   A[0][1] = khalf ? f.r   : f.y;
  A[1][0] = khalf ? f.fcz : f.fcx; A[1][1] = khalf ? f.fnx : f.fcy;
  A[2][0] = khalf ? 0.f   : f.fny; A[2][1] = 0.f;               // K=9..11 pad
}

// ---------------- pass 2: GEMM via WMMA + BN statistics ----------------
__global__ __launch_bounds__(256) void k_bn_stats(
    const float* __restrict__ pts, const int* __restrict__ seg,
    const float* __restrict__ cnt, const float* __restrict__ xsum,
    const float* __restrict__ W, float* __restrict__ bnacc, int n) {
  int lane  = threadIdx.x & 31;
  int nl    = lane & 15;
  int khalf = lane >> 4;
  v2f Bf[3][4];
  load_Bfrag(W, nl, khalf, Bf);

  int wid    = (blockIdx.x * blockDim.x + threadIdx.x) >> 5;
  int nwaves = (gridDim.x * blockDim.x) >> 5;
  int ntiles = (n + 15) >> 4;

  float sAcc[4]  = {0.f, 0.f, 0.f, 0.f};
  float s2Acc[4] = {0.f, 0.f, 0.f, 0.f};

  for (int t = wid; t < ntiles; t += nwaves) {
    int base = t << 4;
    Feats f = compute_feats(pts, seg, cnt, xsum, base + nl, n);
    v2f A[3];
    make_Afrag(f, khalf, A);
    bool fullTile = (base + 16 <= n);          // wave-uniform branch
#pragma unroll
    for (int nt = 0; nt < 4; ++nt) {
      v8f c = {};
#pragma unroll
      for (int kt = 0; kt < 3; ++kt)
        c = __builtin_amdgcn_wmma_f32_16x16x4_f32(
            false, A[kt], false, Bf[kt][nt], (short)0, c, false, false);
      if (fullTile) {                          // fast path: no per-element masks
        float s0 = 0.f, s1 = 0.f;
#pragma unroll
        for (int v = 0; v < 8; ++v) {
          float xv = c[v];
          s0 += xv;
          s1 = fmaf(xv, xv, s1);
        }
        sAcc[nt]  += s0;
        s2Acc[nt] += s1;
      } else {                                 // tail tile only
#pragma unroll
        for (int v = 0; v < 8; ++v) {
          int m = v + 8 * khalf;
          if (base + m < n) {
            float xv = c[v];
            sAcc[nt]  += xv;
            s2Acc[nt] = fmaf(xv, xv, s2Acc[nt]);
          }
        }
      }
    }
  }
#pragma unroll
  for (int nt = 0; nt < 4; ++nt) {
    atomicAdd(&bnacc[       nt * 16 + nl], sAcc[nt]);
    atomicAdd(&bnacc[CCH_ + nt * 16 + nl], s2Acc[nt]);
  }
}

// ---------------- pass 3: fold BN into per-channel scale/shift ----------------
__global__ void k_bn_finalize(const float* __restrict__ bnacc,
                              const float* __restrict__ gamma,
                              const float* __restrict__ beta,
                              float* __restrict__ ss, int n) {
  int c = threadIdx.x;
  if (c < CCH_) {
    float invN = 1.0f / (float)n;
    float mu   = bnacc[c] * invN;
    float var  = bnacc[CCH_ + c] * invN - mu * mu;   // biased var, matches reference
    float sc   = gamma[c] * rsqrtf(var + BN_EPS_);
    ss[c]        = sc;
    ss[CCH_ + c] = beta[c] - mu * sc;
  }
}

// ---------------- pass 4: recompute GEMM, BN+ReLU, scatter-max ----------------
__global__ __launch_bounds__(256) void k_scatter_max(
    const float* __restrict__ pts, const int* __restrict__ seg,
    const float* __restrict__ cnt, const float* __restrict__ xsum,
    const float* __restrict__ W, const float* __restrict__ ss,
    unsigned int* __restrict__ out, int n) {
  int lane  = threadIdx.x & 31;
  int nl    = lane & 15;
  int khalf = lane >> 4;
  v2f Bf[3][4];
  load_Bfrag(W, nl, khalf, Bf);

  float sc[4], sh[4];
#pragma unroll
  for (int nt = 0; nt < 4; ++nt) {
    sc[nt] = ss[       nt * 16 + nl];
    sh[nt] = ss[CCH_ + nt * 16 + nl];
  }

  int wid    = (blockIdx.x * blockDim.x + threadIdx.x) >> 5;
  int nwaves = (gridDim.x * blockDim.x) >> 5;
  int ntiles = (n + 15) >> 4;

  for (int t = wid; t < ntiles; t += nwaves) {
    int base = t << 4;
    Feats f = compute_feats(pts, seg, cnt, xsum, base + nl, n);
    v2f A[3];
    make_Afrag(f, khalf, A);
    int myseg = f.s;                           // seg of point base+nl (NSEG_ if pad)

    // hoist the 8 shuffles (independent of nt): lane m holds seg of point base+m
    int sv[8];
#pragma unroll
    for (int v = 0; v < 8; ++v)
      sv[v] = __shfl(myseg, v + 8 * khalf, 32);

#pragma unroll
    for (int nt = 0; nt < 4; ++nt) {
      v8f c = {};
#pragma unroll
      for (int kt = 0; kt < 3; ++kt)
        c = __builtin_amdgcn_wmma_f32_16x16x4_f32(
            false, A[kt], false, Bf[kt][nt], (short)0, c, false, false);
      int nch = nt * 16 + nl;
#pragma unroll
      for (int v = 0; v < 8; ++v) {
        float y = fmaf(c[v], sc[nt], sh[nt]);  // BN affine; ReLU via >0 gate below
        if (sv[v] < NSEG_ && y > 0.f)          // out pre-zeroed: max(0, y) implicit
          atomicMax(out + ((size_t)sv[v] * CCH_ + nch), __float_as_uint(y));
      }
    }
  }
}

// ---------------- host-side launcher ----------------
extern "C" void kernel_launch(void* const* d_in, const int* in_sizes, int n_in,
                              void* d_out, int out_size, void* d_ws, size_t ws_size,
                              hipStream_t stream) {
  const float* pts   = (const float*)d_in[0];   // [N,5]
  const float* W     = (const float*)d_in[1];   // [9,64]
  const float* gamma = (const float*)d_in[2];   // [64]
  const float* beta  = (const float*)d_in[3];   // [64]
  int n = in_sizes[0] / 5;

  char* ws = (char*)d_ws;
  size_t segOff   = 0;
  size_t segBytes = (((size_t)n * 4) + 255) & ~(size_t)255;
  size_t cntOff   = segOff + segBytes;
  size_t cntBytes = (((size_t)(NSEG_ + 1) * 4) + 255) & ~(size_t)255;
  size_t sumOff   = cntOff + cntBytes;
  size_t sumBytes = (((size_t)(NSEG_ + 1) * 12) + 255) & ~(size_t)255;
  size_t bnOff    = sumOff + sumBytes;
  size_t ssOff    = bnOff + 256;

  int*   seg   = (int*)(ws + segOff);
  float* cnt   = (float*)(ws + cntOff);
  float* xsum  = (float*)(ws + sumOff);
  float* bnacc = (float*)(ws + bnOff);
  float* ss    = (float*)(ws + ssOff);

  // zero pillar stats + BN accumulators (contiguous), and the 256 MB output
  hipMemsetAsync(ws + cntOff, 0, cntBytes + sumBytes + 256 + 2 * CCH_ * sizeof(float), stream);
  hipMemsetAsync(d_out, 0, (size_t)out_size * sizeof(float), stream);
  (void)ws_size; (void)n_in;

  k_pillar_scatter<<<1024, 256, 0, stream>>>(pts, seg, cnt, xsum, n);
  k_bn_stats     <<<1024, 256, 0, stream>>>(pts, seg, cnt, xsum, W, bnacc, n);
  k_bn_finalize  <<<1, 64, 0, stream>>>(bnacc, gamma, beta, ss, n);
  k_scatter_max  <<<1024, 256, 0, stream>>>(pts, seg, cnt, xsum, W, ss,
                                            (unsigned int*)d_out, n);
}